// SemRGCN_80384607912324
// MI455X (gfx1250) — compile-verified
//
#include <hip/hip_runtime.h>

typedef __attribute__((ext_vector_type(2))) float v2f;
typedef __attribute__((ext_vector_type(8))) float v8f;

#define HD 256       // hidden dim
#define NREL 16      // relations
#define NBASES 8     // bases
#define MTILE 128    // M rows per block (8 waves x 16)
#define POFF 260     // padded LDS row stride (floats): 1040 B -> conflict-free b64

// ---------------------------------------------------------------------------
// Zero d_out and the degree counter (harness poisons buffers with 0xAA).
// ---------------------------------------------------------------------------
__global__ __launch_bounds__(256) void zero_kernel(float* __restrict__ out,
                                                   float* __restrict__ cnt,
                                                   int Nn) {
  long idx = (long)blockIdx.x * 256 + threadIdx.x;
  if (idx < (long)Nn * HD) out[idx] = 0.0f;
  if (idx < Nn) cnt[idx] = 0.0f;
}

// ---------------------------------------------------------------------------
// WrelT[r][col][k] = W[r][k][col] = sum_b w_comp[r,b] * basis[b][k][col]
// (transposed relation matrices -> contiguous B-operand fetches later)
// grid: (256 cols, 16 relations), block 256 (thread j = k index; coalesced
// stores of WrelT rows).
// ---------------------------------------------------------------------------
__global__ __launch_bounds__(256) void relw_kernel(const float* __restrict__ basis,
                                                   const float* __restrict__ w_comp,
                                                   float* __restrict__ WrelT) {
  const int j = threadIdx.x;   // k index
  const int i = blockIdx.x;    // column of W (= row of WrelT)
  const int r = blockIdx.y;    // relation
  float acc = 0.0f;
#pragma unroll
  for (int b = 0; b < NBASES; ++b)
    acc = fmaf(w_comp[r * NBASES + b], basis[((size_t)b * HD + j) * HD + i], acc);
  WrelT[((size_t)r * HD + i) * HD + j] = acc;
}

// ---------------------------------------------------------------------------
// Z[r] = X @ W[r] using V_WMMA_F32_16X16X4_F32 (exact f32 math, matrix pipe)
// Block: 256 threads = 8 wave32.  Block tile: M=128, N=256, K=256.
//
// W[r]^T staged in padded LDS via CDNA5 async-to-LDS DMA
// (global_load_async_to_lds_b128, drained by s_wait_asynccnt). Each B operand
// is then ONE ds_load_b64 with an immediate offset: sWT[col*POFF + kb..kb+1].
// Lane stride 1040 B => banks {4l..4l+3} per lane pair-half, conflict-free.
//
// ISA layouts (CDNA5 ISA sec 7.12.2):
//   A 16x4  f32: VGPR0 = {lanes0-15: K=0, lanes16-31: K=2}, VGPR1 = {K=1, K=3},
//                row M = lane & 15.
//   B 4x16  f32: VGPR0 = {K-row 0 | K-row 2} striped over lanes (col = lane&15),
//                VGPR1 = {K-row 1 | K-row 3}.
//   C/D 16x16:   VGPR j = row j (lanes 0-15) / row j+8 (lanes 16-31), col = lane&15.
// ---------------------------------------------------------------------------
__global__ __launch_bounds__(256) void gemm_kernel(const float* __restrict__ emb,
                                                   const int* __restrict__ node_ids,
                                                   const float* __restrict__ WrelT,
                                                   float* __restrict__ Z,
                                                   int Nn) {
  __shared__ float sWT[HD * POFF];  // 256 rows x 260 floats = 266,240 B (< 320 KB)
  const int tid = threadIdx.x;
  const int r = blockIdx.y;

  // --- Async stage of W[r]^T into padded LDS ------------------------------
  // Global side: contiguous float4 stream (coalesced). LDS side: row-padded.
  {
    const unsigned long long gbase =
        (unsigned long long)(WrelT + (size_t)r * HD * HD);
    const unsigned ldsbase = (unsigned)(unsigned long long)(&sWT[0]);
#pragma unroll 1
    for (int q = tid; q < (HD * HD) / 4; q += 256) {   // q in float4 units
      const int col = q >> 6;                          // 64 float4 per row
      const int k4 = q & 63;
      const unsigned lds_addr = ldsbase + (unsigned)(col * (POFF * 4) + k4 * 16);
      const unsigned g_off = (unsigned)(q * 16);
      asm volatile("global_load_async_to_lds_b128 %0, %1, %2"
                   :: "v"(lds_addr), "v"(g_off), "s"(gbase)
                   : "memory");
    }
    asm volatile("s_wait_asynccnt 0x0" ::: "memory");
  }
  __syncthreads();

  const int wave = tid >> 5;
  const int lane = tid & 31;
  const int half = lane >> 4;   // 0 -> K%4 in {0,1}, 1 -> K%4 in {2,3}
  const int lmod = lane & 15;
  const int m0 = blockIdx.x * MTILE + wave * 16;

  // A strip: 16 rows (row = m0 + lane&15), full K=256 -> 64 v2f = 128 VGPRs.
  // Clamp OOB rows (stores are guarded); keeps EXEC all-ones for WMMA.
  const int mrow = m0 + lmod;
  const int mclamp = mrow < Nn ? mrow : (Nn - 1);
  const float* __restrict__ arow = emb + (size_t)node_ids[mclamp] * HD;

  v2f A[64];
#pragma unroll
  for (int s = 0; s < 64; ++s) {
    const float* p = arow + 4 * s + 2 * half;
    A[s] = (v2f){p[0], p[1]};
  }

  float* __restrict__ Zr = Z + (size_t)r * (size_t)Nn * HD;

#pragma unroll 1
  for (int nt = 0; nt < 16; ++nt) {
    const int ncol = nt * 16 + lmod;
    // B fetch base: sWT[ncol*POFF + 2*half], step 4 floats (16 B) per K-step.
    const float* __restrict__ bp = &sWT[ncol * POFF + 2 * half];
    // Two accumulators to break the back-to-back D->C WMMA dependency chain.
    v8f acc0 = (v8f){0.f, 0.f, 0.f, 0.f, 0.f, 0.f, 0.f, 0.f};
    v8f acc1 = (v8f){0.f, 0.f, 0.f, 0.f, 0.f, 0.f, 0.f, 0.f};
#pragma unroll
    for (int s = 0; s < 64; s += 2) {
      {
        const v2f b = *(const v2f*)(bp + 4 * s);            // one ds_load_b64
        acc0 = __builtin_amdgcn_wmma_f32_16x16x4_f32(false, A[s], false, b,
                                                     (short)0, acc0, false, false);
      }
      {
        const v2f b = *(const v2f*)(bp + 4 * (s + 1));      // one ds_load_b64
        acc1 = __builtin_amdgcn_wmma_f32_16x16x4_f32(false, A[s + 1], false, b,
                                                     (short)0, acc1, false, false);
      }
    }
    const int rowbase = m0 + half * 8;
#pragma unroll
    for (int j = 0; j < 8; ++j) {
      const int row = rowbase + j;
      if (row < Nn) Zr[(size_t)row * HD + ncol] = acc0[j] + acc1[j];
    }
  }
}

// ---------------------------------------------------------------------------
// Per-edge message + scatter: out[dst] += norm * Z[etype][src]; cnt[dst] += 1.
// One wave per edge: coalesced 1 KB row read, f32 global atomics.
// ---------------------------------------------------------------------------
__global__ __launch_bounds__(256) void edge_kernel(const int* __restrict__ src,
                                                   const int* __restrict__ dst,
                                                   const int* __restrict__ et,
                                                   const float* __restrict__ norm,
                                                   const float* __restrict__ Z,
                                                   float* __restrict__ out,
                                                   float* __restrict__ cnt,
                                                   int E, int Nn) {
  const int e = blockIdx.x * 8 + (threadIdx.x >> 5);
  if (e >= E) return;
  const int lane = threadIdx.x & 31;
  const int sI = src[e];
  const int dI = dst[e];
  const int rI = et[e];
  const float nm = norm[e];
  const float* __restrict__ zrow = Z + ((size_t)rI * (size_t)Nn + sI) * HD;
  float* __restrict__ orow = out + (size_t)dI * HD;
#pragma unroll
  for (int q = 0; q < 2; ++q) {
    const int base = lane * 8 + q * 4;
    const float4 v = *(const float4*)(&zrow[base]);
    atomicAdd(&orow[base + 0], nm * v.x);
    atomicAdd(&orow[base + 1], nm * v.y);
    atomicAdd(&orow[base + 2], nm * v.z);
    atomicAdd(&orow[base + 3], nm * v.w);
  }
  if (lane == 0) atomicAdd(&cnt[dI], 1.0f);
}

// ---------------------------------------------------------------------------
// Mean + bias:  out = (cnt>0 ? out/cnt : 0) + bias
// ---------------------------------------------------------------------------
__global__ __launch_bounds__(256) void finalize_kernel(float* __restrict__ out,
                                                       const float* __restrict__ cnt,
                                                       const float* __restrict__ bias,
                                                       int Nn) {
  const long idx = (long)blockIdx.x * 256 + threadIdx.x;
  if (idx >= (long)Nn * HD) return;
  const int v = (int)(idx >> 8);
  const int dcol = (int)(idx & (HD - 1));
  const float c = cnt[v];
  const float val = out[idx];
  out[idx] = (c > 0.0f ? val / c : 0.0f) + bias[dcol];
}

// ---------------------------------------------------------------------------
extern "C" void kernel_launch(void* const* d_in, const int* in_sizes, int n_in,
                              void* d_out, int out_size, void* d_ws, size_t ws_size,
                              hipStream_t stream) {
  const int*   node_ids = (const int*)d_in[0];
  const int*   src      = (const int*)d_in[1];
  const int*   dst      = (const int*)d_in[2];
  const int*   et       = (const int*)d_in[3];
  const float* norm     = (const float*)d_in[4];
  const float* emb      = (const float*)d_in[5];
  const float* basis    = (const float*)d_in[6];
  const float* w_comp   = (const float*)d_in[7];
  const float* bias     = (const float*)d_in[8];
  float* out = (float*)d_out;

  const int Nn = in_sizes[0];   // 100000 nodes
  const int E  = in_sizes[1];   // 800000 edges

  // Workspace layout (f32 elements): WrelT[16*256*256] | cnt[Nn] | Z[16*Nn*256]
  float* ws    = (float*)d_ws;
  float* WrelT = ws;
  float* cnt   = WrelT + (size_t)NREL * HD * HD;
  float* Z     = cnt + (((size_t)Nn + 15) & ~(size_t)15);

  const long totOut = (long)Nn * HD;
  const int gridFull = (int)((totOut + 255) / 256);

  zero_kernel<<<gridFull, 256, 0, stream>>>(out, cnt, Nn);
  relw_kernel<<<dim3(HD, NREL), 256, 0, stream>>>(basis, w_comp, WrelT);
  const int mblocks = (Nn + MTILE - 1) / MTILE;
  gemm_kernel<<<dim3(mblocks, NREL), 256, 0, stream>>>(emb, node_ids, WrelT, Z, Nn);
  edge_kernel<<<(E + 7) / 8, 256, 0, stream>>>(src, dst, et, norm, Z, out, cnt, E, Nn);
  finalize_kernel<<<gridFull, 256, 0, stream>>>(out, cnt, bias, Nn);
}